// dgl_sgc_18047452578214
// MI455X (gfx1250) — compile-verified
//
#include <hip/hip_runtime.h>
#include <cstdint>
#include <cstddef>

#define N_NODES 100000
#define N_EDGES 1600000
#define D_IN    128
#define D_HID   128
#define D_OUT   40

typedef __attribute__((ext_vector_type(2))) float v2f;
typedef __attribute__((ext_vector_type(8))) float v8f;

// ---------------------------------------------------------------- utilities
__global__ __launch_bounds__(256) void zero_u32_kernel(unsigned* __restrict__ p, int n) {
    int i = blockIdx.x * blockDim.x + threadIdx.x;
    if (i < n) p[i] = 0u;
}

__global__ __launch_bounds__(256) void deg_kernel(const int* __restrict__ dst,
                                                  unsigned* __restrict__ deg) {
    int e = blockIdx.x * blockDim.x + threadIdx.x;
    if (e < N_EDGES) atomicAdd(&deg[dst[e]], 1u);
}

__global__ __launch_bounds__(256) void norm_kernel(const unsigned* __restrict__ deg,
                                                   float* __restrict__ nrm) {
    int i = blockIdx.x * blockDim.x + threadIdx.x;
    if (i < N_NODES) {
        float d = fmaxf((float)deg[i], 1.0f);
        nrm[i] = rsqrtf(d);
    }
}

// -------------------------------------------------- edge gather/scatter-add
// One 32-lane wave per edge; each lane handles dims {lane, lane+32, lane+64,
// lane+96} -> fully coalesced 128B segments, f32 atomics resolve in L2
// (entire agg buffer = 51 MB << 192 MB L2).
__global__ __launch_bounds__(256) void scatter_kernel(const float* __restrict__ x,
                                                      const float* __restrict__ nrm,
                                                      const int* __restrict__ src,
                                                      const int* __restrict__ dst,
                                                      float* __restrict__ agg) {
    long long gid = (long long)blockIdx.x * blockDim.x + threadIdx.x;
    int e    = (int)(gid >> 5);
    int lane = (int)(gid & 31);
    if (e >= N_EDGES) return;
    int s = src[e];
    int d = dst[e];
    float ns = nrm[s];                       // fuse src-side norm into gather
    const float* xp = x   + (size_t)s * D_IN;
    float*       ap = agg + (size_t)d * D_IN;
#pragma unroll
    for (int j = 0; j < 4; ++j) {
        int k = lane + j * 32;
        unsafeAtomicAdd(&ap[k], xp[k] * ns); // global_atomic_add_f32, no return
    }
}

// ---------------------------------------------------------- WMMA GEMM no.1
// h1 = elu( (agg * norm_row) @ W1 + b1 )   [100000 x 128] x [128 x 128]
// Block = 256 thr (8 waves). Block owns 16 rows; wave w owns cols [16w,16w+16).
// fp32 WMMA: V_WMMA_F32_16X16X4_F32, K-loop of 32 steps.
#define LDS_STRIDE 132   // 128 + 4: (row*132 + k) % 64 unique per row -> no bank conflicts

__global__ __launch_bounds__(256) void sgc_gemm1_kernel(const float* __restrict__ agg,
                                                        const float* __restrict__ nrm,
                                                        const float* __restrict__ W1,
                                                        const float* __restrict__ b1,
                                                        float* __restrict__ h1) {
    __shared__ float As[16 * LDS_STRIDE];
    const int m0  = blockIdx.x * 16;          // 6250 * 16 == 100000 exactly
    const int tid = threadIdx.x;

    // Stage A tile (16 x 128), fused dst-side row scaling by norm[m].
    for (int i = tid; i < 16 * 128; i += 256) {
        int r = i >> 7, c = i & 127;
        As[r * LDS_STRIDE + c] = agg[(size_t)(m0 + r) * D_IN + c] * nrm[m0 + r];
    }
    __syncthreads();

    const int wave = tid >> 5;        // 0..7 -> N tile
    const int lane = tid & 31;
    const int half = lane >> 4;       // 0: K pair {0,1}, 1: K pair {2,3}
    const int l16  = lane & 15;
    const int n0   = wave * 16;
    const int koff = half * 2;

    v8f acc = {};
#pragma unroll 4
    for (int kk = 0; kk < D_IN; kk += 4) {
        v2f a, b;
        // A 16x4 layout: lanes 0-15 hold (M=lane, K=kk..kk+1), lanes 16-31 K=kk+2..kk+3
        a.x = As[l16 * LDS_STRIDE + kk + koff];
        a.y = As[l16 * LDS_STRIDE + kk + koff + 1];
        // B 4x16 layout mirrored: row K striped across lanes (N = n0+l16)
        b.x = W1[(size_t)(kk + koff)     * D_HID + n0 + l16];
        b.y = W1[(size_t)(kk + koff + 1) * D_HID + n0 + l16];
        acc = __builtin_amdgcn_wmma_f32_16x16x4_f32(false, a, false, b,
                                                    (short)0, acc, false, false);
    }

    const float bias = b1[n0 + l16];
#pragma unroll
    for (int r = 0; r < 8; ++r) {
        // C/D layout: VGPR r -> M = r (lanes 0-15) or r+8 (lanes 16-31)
        int   m = m0 + r + half * 8;
        float v = acc[r] + bias;
        v = (v > 0.0f) ? v : expm1f(v);                   // ELU, alpha=1
        h1[(size_t)m * D_HID + n0 + l16] = v;             // un-normed; norm fused at next gather
    }
}

// ---------------------------------------------------------- WMMA GEMM no.2
// out = (agg * norm_row) @ W2 + b2   [100000 x 128] x [128 x 40]
// Block = 96 thr (3 waves) -> N tiles {0..15},{16..31},{32..47}, masked to 40.
__global__ __launch_bounds__(96) void sgc_gemm2_kernel(const float* __restrict__ agg,
                                                       const float* __restrict__ nrm,
                                                       const float* __restrict__ W2,
                                                       const float* __restrict__ b2,
                                                       float* __restrict__ out) {
    __shared__ float As[16 * LDS_STRIDE];
    const int m0  = blockIdx.x * 16;
    const int tid = threadIdx.x;

    for (int i = tid; i < 16 * 128; i += 96) {
        int r = i >> 7, c = i & 127;
        As[r * LDS_STRIDE + c] = agg[(size_t)(m0 + r) * D_HID + c] * nrm[m0 + r];
    }
    __syncthreads();

    const int wave = tid >> 5;        // 0..2
    const int lane = tid & 31;
    const int half = lane >> 4;
    const int l16  = lane & 15;
    const int n    = wave * 16 + l16;
    const bool nok = (n < D_OUT);
    const int koff = half * 2;

    v8f acc = {};
#pragma unroll 4
    for (int kk = 0; kk < D_HID; kk += 4) {
        v2f a, b;
        a.x = As[l16 * LDS_STRIDE + kk + koff];
        a.y = As[l16 * LDS_STRIDE + kk + koff + 1];
        b.x = nok ? W2[(size_t)(kk + koff)     * D_OUT + n] : 0.0f;
        b.y = nok ? W2[(size_t)(kk + koff + 1) * D_OUT + n] : 0.0f;
        acc = __builtin_amdgcn_wmma_f32_16x16x4_f32(false, a, false, b,
                                                    (short)0, acc, false, false);
    }

    const float bias = nok ? b2[n] : 0.0f;
#pragma unroll
    for (int r = 0; r < 8; ++r) {
        if (nok) {
            int m = m0 + r + half * 8;
            out[(size_t)m * D_OUT + n] = acc[r] + bias;
        }
    }
}

// ------------------------------------------------------------------ driver
extern "C" void kernel_launch(void* const* d_in, const int* in_sizes, int n_in,
                              void* d_out, int out_size, void* d_ws, size_t ws_size,
                              hipStream_t stream) {
    const float* features = (const float*)d_in[0];
    const int*   src      = (const int*)  d_in[1];
    const int*   dst      = (const int*)  d_in[2];
    const float* W1       = (const float*)d_in[3];
    const float* b1       = (const float*)d_in[4];
    const float* W2       = (const float*)d_in[5];
    const float* b2       = (const float*)d_in[6];
    float*       out      = (float*)d_out;

    // Workspace carve-out (~103 MB): deg | norm | agg | h1
    size_t off = 0;
    auto carve = [&](size_t nbytes) {
        void* p = (char*)d_ws + off;
        off += (nbytes + 255) & ~(size_t)255;
        return p;
    };
    unsigned* deg = (unsigned*)carve((size_t)N_NODES * sizeof(unsigned));
    float*    nrm = (float*)   carve((size_t)N_NODES * sizeof(float));
    float*    agg = (float*)   carve((size_t)N_NODES * D_HID * sizeof(float));
    float*    h1  = (float*)   carve((size_t)N_NODES * D_HID * sizeof(float));

    const int aggWords = N_NODES * D_HID;

    // degrees + norm
    zero_u32_kernel<<<(N_NODES + 255) / 256, 256, 0, stream>>>(deg, N_NODES);
    deg_kernel<<<(N_EDGES + 255) / 256, 256, 0, stream>>>(dst, deg);
    norm_kernel<<<(N_NODES + 255) / 256, 256, 0, stream>>>(deg, nrm);

    const long long scatterThreads = (long long)N_EDGES * 32;
    const int scatterBlocks = (int)((scatterThreads + 255) / 256);

    // ---- layer 1 ----
    zero_u32_kernel<<<(aggWords + 255) / 256, 256, 0, stream>>>((unsigned*)agg, aggWords);
    scatter_kernel<<<scatterBlocks, 256, 0, stream>>>(features, nrm, src, dst, agg);
    sgc_gemm1_kernel<<<N_NODES / 16, 256, 0, stream>>>(agg, nrm, W1, b1, h1);

    // ---- layer 2 ----
    zero_u32_kernel<<<(aggWords + 255) / 256, 256, 0, stream>>>((unsigned*)agg, aggWords);
    scatter_kernel<<<scatterBlocks, 256, 0, stream>>>(h1, nrm, src, dst, agg);
    sgc_gemm2_kernel<<<N_NODES / 16, 96, 0, stream>>>(agg, nrm, W2, b2, out);
}